// PostProcess_48859547959902
// MI455X (gfx1250) — compile-verified
//
#include <hip/hip_runtime.h>
#include <hip/hip_bf16.h>
#include <math.h>
#include <stdint.h>

#define TPB   256
#define BATCH 128
#define QN    2000
#define CN    91
#define KTOP  100
#define NTOT  (QN * CN)   /* 182000 */
#define NV4   (NTOT / 4)  /* 45500  */
#define BINS  4096
#define CAP   4096

// ---- order-preserving float <-> u32 ----
__device__ __forceinline__ unsigned ordf(float f) {
  unsigned u = __float_as_uint(f);
  return (u & 0x80000000u) ? ~u : (u | 0x80000000u);
}
__device__ __forceinline__ float unordf(unsigned u) {
  unsigned b = (u & 0x80000000u) ? (u & 0x7FFFFFFFu) : ~u;
  return __uint_as_float(b);
}
__device__ __forceinline__ float sigmoidf(float x) {
  return 1.0f / (1.0f + __expf(-x));
}

// ---- CDNA5 async global->LDS (ASYNCcnt-tracked) ----
__device__ __forceinline__ void async_g2l_b128(unsigned lds_byte_off, const void* gaddr) {
  asm volatile("global_load_async_to_lds_b128 %0, %1, off"
               :: "v"(lds_byte_off), "v"((unsigned long long)(uintptr_t)gaddr)
               : "memory");
}
__device__ __forceinline__ void wait_asynccnt_le1() {
  asm volatile("s_wait_asynccnt 1" ::: "memory");
}
__device__ __forceinline__ void wait_asynccnt_0() {
  asm volatile("s_wait_asynccnt 0" ::: "memory");
}
__device__ __forceinline__ void wait_dscnt_0() {
  asm volatile("s_wait_dscnt 0" ::: "memory");
}

__global__ __launch_bounds__(TPB)
void detr_postprocess_kernel(const float* __restrict__ logits,
                             const float* __restrict__ boxes,
                             const int*   __restrict__ tsizes,
                             float*       __restrict__ out) {
  __shared__ unsigned           hist[BINS];        // 16 KB
  __shared__ unsigned long long keys[CAP];         // 32 KB
  __shared__ float4             stage[2][TPB];     //  8 KB
  __shared__ unsigned           gsum[TPB];         //  1 KB (16-bin group sums)
  __shared__ int                s_tbin;
  __shared__ unsigned           s_cnt;
  __shared__ int                topq[KTOP];

  const int b   = blockIdx.x;
  const int tid = threadIdx.x;
  const float*  blogits = logits + (size_t)b * NTOT;
  const float4* src4    = (const float4*)blogits;

  for (int i = tid; i < BINS; i += TPB) hist[i] = 0u;
  if (tid == 0) s_cnt = 0u;
  __syncthreads();

  // ---------- Pass 1: histogram of top-12 ordered bits, double-buffered async staging ----------
  const int iters = (NV4 + TPB - 1) / TPB;  // 178
  {
    int e = tid; if (e >= NV4) e = NV4 - 1;
    async_g2l_b128((unsigned)(uintptr_t)&stage[0][tid], src4 + e);
  }
  int p = 0;
  for (int it = 0; it < iters; ++it) {
    if (it + 1 < iters) {
      int e = (it + 1) * TPB + tid; if (e >= NV4) e = NV4 - 1;
      async_g2l_b128((unsigned)(uintptr_t)&stage[p ^ 1][tid], src4 + e);
      wait_asynccnt_le1();   // oldest async (current buffer) is complete
    } else {
      wait_asynccnt_0();
    }
    float4 v = stage[p][tid];
    wait_dscnt_0();          // readback in regs before next iter's async overwrite
    int e = it * TPB + tid;
    if (e < NV4) {
      atomicAdd(&hist[ordf(v.x) >> 20], 1u);
      atomicAdd(&hist[ordf(v.y) >> 20], 1u);
      atomicAdd(&hist[ordf(v.z) >> 20], 1u);
      atomicAdd(&hist[ordf(v.w) >> 20], 1u);
    }
    p ^= 1;
  }
  __syncthreads();

  // ---------- Threshold: two-level suffix scan (256 groups of 16 bins) ----------
  {
    unsigned s = 0;
    #pragma unroll
    for (int j = 0; j < 16; ++j) s += hist[tid * 16 + j];
    gsum[tid] = s;
  }
  __syncthreads();
  if (tid == 0) {
    unsigned cum = 0; int g = TPB - 1;
    for (; g > 0; --g) { cum += gsum[g]; if (cum >= KTOP) break; }
    if (cum < KTOP) cum += gsum[0], g = 0;      // g==0 fallthrough
    unsigned above = cum - gsum[g];             // count in bins strictly above group g
    int t = g * 16 + 15;
    unsigned cum2 = above;
    for (; t > g * 16; --t) { cum2 += hist[t]; if (cum2 >= KTOP) break; }
    s_tbin = t;                                 // suffix(t) >= KTOP, suffix(t+1) < KTOP
  }
  __syncthreads();
  const unsigned tbin = (unsigned)s_tbin;

  // ---------- Pass 2: collect candidate (value, ~index) keys ----------
  for (int e = tid; e < NV4; e += TPB) {
    if (e + TPB < NV4) __builtin_prefetch(src4 + e + TPB, 0, 1);  // global_prefetch_b8
    float4 v = src4[e];
    unsigned u0 = ordf(v.x), u1 = ordf(v.y), u2 = ordf(v.z), u3 = ordf(v.w);
    unsigned i0 = 4u * (unsigned)e;
    if ((u0 >> 20) >= tbin) { unsigned pos = atomicAdd(&s_cnt, 1u);
      if (pos < CAP) keys[pos] = ((unsigned long long)u0 << 32) | (unsigned)~(i0 + 0u); }
    if ((u1 >> 20) >= tbin) { unsigned pos = atomicAdd(&s_cnt, 1u);
      if (pos < CAP) keys[pos] = ((unsigned long long)u1 << 32) | (unsigned)~(i0 + 1u); }
    if ((u2 >> 20) >= tbin) { unsigned pos = atomicAdd(&s_cnt, 1u);
      if (pos < CAP) keys[pos] = ((unsigned long long)u2 << 32) | (unsigned)~(i0 + 2u); }
    if ((u3 >> 20) >= tbin) { unsigned pos = atomicAdd(&s_cnt, 1u);
      if (pos < CAP) keys[pos] = ((unsigned long long)u3 << 32) | (unsigned)~(i0 + 3u); }
  }
  __syncthreads();

  // ---------- Sort size: smallest pow2 >= cnt, in [128, CAP] ----------
  unsigned cnt = s_cnt; if (cnt > CAP) cnt = CAP;
  int n = 128;
  while ((unsigned)n < cnt) n <<= 1;           // n <= CAP since cnt <= CAP
  for (int i = (int)cnt + tid; i < n; i += TPB) keys[i] = 0ull;
  __syncthreads();

  // ---------- Bitonic sort, descending (key = value:~index => jax tie-break) ----------
  for (int k = 2; k <= n; k <<= 1) {
    for (int j = k >> 1; j > 0; j >>= 1) {
      for (int t = tid; t < n; t += TPB) {
        int ixj = t ^ j;
        if (ixj > t) {
          unsigned long long a = keys[t], c = keys[ixj];
          bool desc = ((t & k) == 0);
          if (desc ? (a < c) : (a > c)) { keys[t] = c; keys[ixj] = a; }
        }
      }
      __syncthreads();
    }
  }

  // ---------- Emit outputs: scores | labels | boxes | prob_g (flat, in return order) ----------
  float* out_scores = out;
  float* out_labels = out + (size_t)BATCH * KTOP;
  float* out_boxes  = out + (size_t)2 * BATCH * KTOP;
  float* out_prob   = out + (size_t)2 * BATCH * KTOP + (size_t)BATCH * KTOP * 4;

  const float img_h = (float)tsizes[2 * b + 0];
  const float img_w = (float)tsizes[2 * b + 1];

  if (tid < KTOP) {
    unsigned long long kk = keys[tid];
    unsigned u   = (unsigned)(kk >> 32);
    unsigned idx = ~(unsigned)(kk & 0xFFFFFFFFull);
    unsigned q   = idx / CN;
    unsigned c   = idx % CN;
    float lg = unordf(u);
    out_scores[(size_t)b * KTOP + tid] = sigmoidf(lg);
    out_labels[(size_t)b * KTOP + tid] = (float)c;
    topq[tid] = (int)q;

    float4 bx = ((const float4*)boxes)[(size_t)b * QN + q];
    float x0 = bx.x - 0.5f * bx.z, y0 = bx.y - 0.5f * bx.w;
    float x1 = bx.x + 0.5f * bx.z, y1 = bx.y + 0.5f * bx.w;
    float* ob = out_boxes + ((size_t)b * KTOP + tid) * 4;
    ob[0] = x0 * img_w; ob[1] = y0 * img_h;
    ob[2] = x1 * img_w; ob[3] = y1 * img_h;
  }
  __syncthreads();

  for (int t = tid; t < KTOP * CN; t += TPB) {
    int r = t / CN, c = t % CN;
    int q = topq[r];
    out_prob[(size_t)b * KTOP * CN + t] = sigmoidf(blogits[(size_t)q * CN + c]);
  }
}

extern "C" void kernel_launch(void* const* d_in, const int* in_sizes, int n_in,
                              void* d_out, int out_size, void* d_ws, size_t ws_size,
                              hipStream_t stream) {
  const float* logits = (const float*)d_in[0];
  const float* bx     = (const float*)d_in[1];
  const int*   tsz    = (const int*)d_in[2];
  float* out = (float*)d_out;
  (void)in_sizes; (void)n_in; (void)out_size; (void)d_ws; (void)ws_size;
  hipLaunchKernelGGL(detr_postprocess_kernel, dim3(BATCH), dim3(TPB), 0, stream,
                     logits, bx, tsz, out);
}